// simpleGAT_23081154249038
// MI455X (gfx1250) — compile-verified
//
#include <hip/hip_runtime.h>

#define NN      30000
#define NE      480000
#define NEP     (NE + NN)        // edges + self loops = 510000
#define HC      256
#define NHEADS  8
#define NGRAPH  64
#define NEG_SLOPE 0.2f

typedef float v2f __attribute__((ext_vector_type(2)));
typedef float v8f __attribute__((ext_vector_type(8)));

// ---------------------------------------------------------------------------
// float atomic max via sign-split integer atomics (works with -inf init)
__device__ __forceinline__ void atomicMaxF(float* addr, float v) {
  if (v >= 0.0f) atomicMax((int*)addr, __float_as_int(v));
  else           atomicMin((unsigned int*)addr, __float_as_uint(v));
}

__device__ __forceinline__ void edge_ends(const int* __restrict__ ei, int e,
                                          int& s, int& d) {
  if (e < NE) { s = ei[e]; d = ei[NE + e]; }
  else        { s = d = e - NE; }          // self loop
}

// ---------------------------------------------------------------------------
// H[30000,256] = X[30000,K] @ W[K,256]   via V_WMMA_F32_16X16X4_F32 (fp32)
// One wave computes a 16x64 strip: A-fragment reused across 4 N-tiles.
__global__ void __launch_bounds__(32)
k_gemm_wmma(const float* __restrict__ X, const float* __restrict__ W,
            float* __restrict__ H, int K) {
  const int tm   = blockIdx.x;        // 16-row tile (0..1874)
  const int tn   = blockIdx.y;        // 64-col group (0..3)
  const int lane = threadIdx.x;       // wave32
  const int mr   = lane & 15;
  const int kh   = lane >> 4;         // K half-select (lanes 16-31 hold k+2)
  const float* __restrict__ xrow = X + (size_t)(tm * 16 + mr) * (size_t)K;
  const int nbase = tn * 64 + mr;
  v8f acc0 = {}, acc1 = {}, acc2 = {}, acc3 = {};
  for (int k0 = 0; k0 < K; k0 += 4) {
    const int ka = k0 + 2 * kh;
    v2f a; a.x = xrow[ka]; a.y = xrow[ka + 1];          // A[m][ka], A[m][ka+1]
    const float* __restrict__ wrow = W + (size_t)ka * HC + nbase;
    v2f b;
    b.x = wrow[0];        b.y = wrow[HC];
    acc0 = __builtin_amdgcn_wmma_f32_16x16x4_f32(false, a, false, b, (short)0, acc0, false, false);
    b.x = wrow[16];       b.y = wrow[HC + 16];
    acc1 = __builtin_amdgcn_wmma_f32_16x16x4_f32(false, a, false, b, (short)0, acc1, false, false);
    b.x = wrow[32];       b.y = wrow[HC + 32];
    acc2 = __builtin_amdgcn_wmma_f32_16x16x4_f32(false, a, false, b, (short)0, acc2, false, false);
    b.x = wrow[48];       b.y = wrow[HC + 48];
    acc3 = __builtin_amdgcn_wmma_f32_16x16x4_f32(false, a, false, b, (short)0, acc3, false, false);
  }
  // D layout: VGPR v -> row v + 8*kh, col = lane%16
#pragma unroll
  for (int v = 0; v < 8; ++v) {
    float* hr = H + (size_t)(tm * 16 + v + 8 * kh) * HC + tn * 64 + mr;
    hr[0]  = acc0[v];
    hr[16] = acc1[v];
    hr[32] = acc2[v];
    hr[48] = acc3[v];
  }
}

// ---------------------------------------------------------------------------
// a_s[n,h] = <h[n,h,:], att_src[h,:]> ; a_d likewise. One thread per (node,head).
__global__ void k_scores(const float* __restrict__ H,
                         const float* __restrict__ attS,
                         const float* __restrict__ attD,
                         float* __restrict__ aS, float* __restrict__ aD) {
  int t = blockIdx.x * blockDim.x + threadIdx.x;
  if (t >= NN * NHEADS) return;
  int n = t >> 3, hd = t & 7;
  const float* hp = H + (size_t)n * HC + hd * 32;
  const float* ws = attS + hd * 32;
  const float* wd = attD + hd * 32;
  float s = 0.f, d = 0.f;
#pragma unroll
  for (int c = 0; c < 32; ++c) { float v = hp[c]; s += v * ws[c]; d += v * wd[c]; }
  aS[t] = s; aD[t] = d;
}

// ---------------------------------------------------------------------------
__global__ void k_init_layer(float* __restrict__ mB, float* __restrict__ dB,
                             float* __restrict__ acc) {
  const int total = NN * HC;
  for (int i = blockIdx.x * blockDim.x + threadIdx.x; i < total;
       i += gridDim.x * blockDim.x) {
    acc[i] = 0.f;
    if (i < NN * NHEADS) { mB[i] = -__builtin_huge_valf(); dB[i] = 0.f; }
  }
}

// pass 1: segment max of leaky-relu scores
__global__ void k_edge_max(const int* __restrict__ ei, const float* __restrict__ aS,
                           const float* __restrict__ aD, float* __restrict__ mB) {
  int t = blockIdx.x * blockDim.x + threadIdx.x;
  if (t >= NEP * NHEADS) return;
  int e = t >> 3, hd = t & 7;
  int s, d; edge_ends(ei, e, s, d);
  float v = aS[s * NHEADS + hd] + aD[d * NHEADS + hd];
  v = (v > 0.f) ? v : NEG_SLOPE * v;
  atomicMaxF(&mB[d * NHEADS + hd], v);
}

// pass 2: denom = segment sum of exp(e - m)
__global__ void k_edge_sum(const int* __restrict__ ei, const float* __restrict__ aS,
                           const float* __restrict__ aD, const float* __restrict__ mB,
                           float* __restrict__ dB) {
  int t = blockIdx.x * blockDim.x + threadIdx.x;
  if (t >= NEP * NHEADS) return;
  int e = t >> 3, hd = t & 7;
  int s, d; edge_ends(ei, e, s, d);
  float v = aS[s * NHEADS + hd] + aD[d * NHEADS + hd];
  v = (v > 0.f) ? v : NEG_SLOPE * v;
  atomicAdd(&dB[d * NHEADS + hd], __expf(v - mB[d * NHEADS + hd]));
}

// pass 3: acc[dst] += h[src] * alpha. One wave per edge, 8 channels per lane
// (all 8 within one head, so one alpha per lane).
__global__ void __launch_bounds__(256)
k_edge_agg(const int* __restrict__ ei, const float* __restrict__ aS,
           const float* __restrict__ aD, const float* __restrict__ mB,
           const float* __restrict__ dB, const float* __restrict__ H,
           float* __restrict__ acc) {
  int w = (blockIdx.x * blockDim.x + threadIdx.x) >> 5;   // edge id (wave-uniform)
  if (w >= NEP) return;
  int lane = threadIdx.x & 31;
  int s, d; edge_ends(ei, w, s, d);
  int hd = lane >> 2;
  float v = aS[s * NHEADS + hd] + aD[d * NHEADS + hd];
  v = (v > 0.f) ? v : NEG_SLOPE * v;
  float alpha = __expf(v - mB[d * NHEADS + hd]) / dB[d * NHEADS + hd];
  const float4* hp = (const float4*)(H + (size_t)s * HC + lane * 8);
  float4 h0 = hp[0], h1 = hp[1];
  float* op = acc + (size_t)d * HC + lane * 8;
  atomicAdd(op + 0, h0.x * alpha); atomicAdd(op + 1, h0.y * alpha);
  atomicAdd(op + 2, h0.z * alpha); atomicAdd(op + 3, h0.w * alpha);
  atomicAdd(op + 4, h1.x * alpha); atomicAdd(op + 5, h1.y * alpha);
  atomicAdd(op + 6, h1.z * alpha); atomicAdd(op + 7, h1.w * alpha);
}

// out = relu(acc + bias) -> next layer input
__global__ void k_finalize(const float* __restrict__ acc, const float* __restrict__ b,
                           float* __restrict__ X) {
  int i = blockIdx.x * blockDim.x + threadIdx.x;
  if (i >= NN * HC) return;
  X[i] = fmaxf(acc[i] + b[i & (HC - 1)], 0.f);
}

// ---------------------------------------------------------------------------
__global__ void k_pool_init(float* __restrict__ pooled, float* __restrict__ cnt) {
  int i = blockIdx.x * blockDim.x + threadIdx.x;
  if (i < NGRAPH * HC) pooled[i] = 0.f;
  if (i < NGRAPH) cnt[i] = 0.f;
}

__global__ void __launch_bounds__(256)
k_pool(const float* __restrict__ X, const int* __restrict__ batch,
       float* __restrict__ pooled, float* __restrict__ cnt) {
  int n = blockIdx.x, c = threadIdx.x;
  int g = batch[n];
  atomicAdd(&pooled[(size_t)g * HC + c], X[(size_t)n * HC + c]);
  if (c == 0) atomicAdd(&cnt[g], 1.0f);
}

__global__ void __launch_bounds__(128)
k_mlp(const float* __restrict__ pooled, const float* __restrict__ cnt,
      const float* __restrict__ l1w, const float* __restrict__ l1b,
      const float* __restrict__ l2w, const float* __restrict__ l2b,
      float* __restrict__ out) {
  __shared__ float pl[HC];
  __shared__ float hid[80];
  int g = blockIdx.x, t = threadIdx.x;
  float inv = 1.0f / fmaxf(cnt[g], 1.0f);
  for (int i = t; i < HC; i += 128) pl[i] = pooled[g * HC + i] * inv;
  __syncthreads();
  if (t < 80) {
    float s = l1b[t];
    for (int i = 0; i < HC; ++i) s += pl[i] * l1w[i * 80 + t];
    hid[t] = fmaxf(s, 0.f);
  }
  __syncthreads();
  if (t < 16) {
    float s = l2b[t];
    for (int i = 0; i < 80; ++i) s += hid[i] * l2w[i * 16 + t];
    out[g * 16 + t] = s;
  }
}

// ---------------------------------------------------------------------------
extern "C" void kernel_launch(void* const* d_in, const int* in_sizes, int n_in,
                              void* d_out, int out_size, void* d_ws, size_t ws_size,
                              hipStream_t stream) {
  (void)in_sizes; (void)n_in; (void)out_size; (void)ws_size;
  const float* x     = (const float*)d_in[0];
  const int*   ei    = (const int*)d_in[1];
  const int*   batch = (const int*)d_in[2];
  const float* Wm[3]   = {(const float*)d_in[3],  (const float*)d_in[7],  (const float*)d_in[11]};
  const float* as[3]   = {(const float*)d_in[4],  (const float*)d_in[8],  (const float*)d_in[12]};
  const float* ad[3]   = {(const float*)d_in[5],  (const float*)d_in[9],  (const float*)d_in[13]};
  const float* bias[3] = {(const float*)d_in[6],  (const float*)d_in[10], (const float*)d_in[14]};
  const float* l1w = (const float*)d_in[15];
  const float* l1b = (const float*)d_in[16];
  const float* l2w = (const float*)d_in[17];
  const float* l2b = (const float*)d_in[18];
  float* out = (float*)d_out;

  // workspace carve-up (floats): 3x feature buffers + score/softmax buffers
  float* B0 = (float*)d_ws;                       // layer input (after finalize)
  float* B1 = B0 + (size_t)NN * HC;               // GEMM output h
  float* B2 = B1 + (size_t)NN * HC;               // aggregation accumulator
  float* aS = B2 + (size_t)NN * HC;               // [NN,8]
  float* aD = aS + (size_t)NN * NHEADS;
  float* mB = aD + (size_t)NN * NHEADS;
  float* dB = mB + (size_t)NN * NHEADS;
  float* pooled = dB + (size_t)NN * NHEADS;       // [64,256]
  float* cnt    = pooled + NGRAPH * HC;           // [64]

  const int eh_threads = NEP * NHEADS;
  const int eh_blocks  = (eh_threads + 255) / 256;

  const float* in = x;
  for (int L = 0; L < 3; ++L) {
    const int K = (L == 0) ? 128 : HC;
    k_gemm_wmma<<<dim3(NN / 16, 4), 32, 0, stream>>>(in, Wm[L], B1, K);
    k_scores<<<(NN * NHEADS + 255) / 256, 256, 0, stream>>>(B1, as[L], ad[L], aS, aD);
    k_init_layer<<<1024, 256, 0, stream>>>(mB, dB, B2);
    k_edge_max<<<eh_blocks, 256, 0, stream>>>(ei, aS, aD, mB);
    k_edge_sum<<<eh_blocks, 256, 0, stream>>>(ei, aS, aD, mB, dB);
    k_edge_agg<<<(NEP + 7) / 8, 256, 0, stream>>>(ei, aS, aD, mB, dB, B1, B2);
    k_finalize<<<(NN * HC + 255) / 256, 256, 0, stream>>>(B2, bias[L], B0);
    in = B0;
  }

  k_pool_init<<<(NGRAPH * HC + 255) / 256, 256, 0, stream>>>(pooled, cnt);
  k_pool<<<NN, HC, 0, stream>>>(B0, batch, pooled, cnt);
  k_mlp<<<NGRAPH, 128, 0, stream>>>(pooled, cnt, l1w, l1b, l2w, l2b, out);
}